// minerva_base_10969346474460
// MI455X (gfx1250) — compile-verified
//
#include <hip/hip_runtime.h>
#include <hip/hip_bf16.h>
#include <math.h>

typedef __attribute__((ext_vector_type(2))) float v2f;
typedef __attribute__((ext_vector_type(8))) float v8f;

#define LDT   36    // LDS row stride (floats) for 32-wide K tiles: 144B (16B aligned), conflict-free
#define LDA_S 132   // LDS row stride for 128-wide a-tile: 528B (16B aligned), conflict-free
#define NSPLIT 8

__device__ __forceinline__ v8f v8f_zero() {
  v8f z; for (int i = 0; i < 8; ++i) z[i] = 0.f; return z;
}

__device__ __forceinline__ v8f wmma_f32(v2f a, v2f b, v8f c) {
  // D = A(16x4,f32) * B(4x16,f32) + C(16x16,f32)
  return __builtin_amdgcn_wmma_f32_16x16x4_f32(false, a, false, b, (short)0, c, false, false);
}

// CDNA5 async global->LDS copy (16B per lane, tracked by ASYNCcnt; data does
// not pass through VGPRs, so no s_wait_loadcnt serialization).
__device__ __forceinline__ void async_ld_b128(void* lds, const void* g) {
  uint32_t l = (uint32_t)(uintptr_t)lds;   // low 32 bits of flat addr == LDS byte offset
  asm volatile("global_load_async_to_lds_b128 %0, %1, off"
               :: "v"(l), "v"(g) : "memory");
}

__device__ __forceinline__ void wait_async0() {
#if __has_builtin(__builtin_amdgcn_s_wait_asynccnt)
  __builtin_amdgcn_s_wait_asynccnt(0);
#else
  asm volatile("s_wait_asynccnt 0x0" ::: "memory");
#endif
}

// Stage a 128x32 fp32 chunk (rows row0.., cols kk..kk+32 of a K-contiguous
// matrix) into LDS tile with row stride LDT, via async loads. 256 threads.
__device__ __forceinline__ void stage_async(const float* __restrict__ g, size_t ld,
                                            int row0, int kk, float* lds, int tid) {
  for (int i = 0; i < 4; ++i) {
    int s = tid + i * 256;
    int r = s >> 3, c = (s & 7) << 2;
    async_ld_b128(&lds[r * LDT + c], &g[(size_t)(row0 + r) * ld + kk + c]);
  }
}

// ---------------------------------------------------------------------------
// C[m][n] = sum_k A[m][k]*B[n][k] + bias[n]   (both operands row-major, K-contig)
// 128x128 tiles, BK=32 double-buffered via async-LDS, 8 waves as 4(M)x2(N).
// ---------------------------------------------------------------------------
__global__ __launch_bounds__(256)
void gemm_nt_bias_kernel(const float* __restrict__ A, const float* __restrict__ B,
                         const float* __restrict__ bias, float* __restrict__ C,
                         int M, int N, int K) {
  __shared__ float As[2 * 128 * LDT];
  __shared__ float Bs[2 * 128 * LDT];

  const int tid  = threadIdx.x;
  const int lane = tid & 31;
  const int w    = tid >> 5;
  const int wm   = w & 3;
  const int wn   = w >> 2;
  const int lrow = lane & 15;
  const int koff = (lane >> 4) << 1;
  const int mhalf = (lane >> 4) << 3;

  const int m0 = blockIdx.y * 128;
  const int n0 = blockIdx.x * 128;

  v8f acc[2][4];
  for (int i = 0; i < 2; ++i) for (int j = 0; j < 4; ++j) acc[i][j] = v8f_zero();

  const int nk = K >> 5;
  stage_async(A, K, m0, 0, &As[0], tid);
  stage_async(B, K, n0, 0, &Bs[0], tid);
  wait_async0();
  __syncthreads();

  for (int kki = 0; kki < nk; ++kki) {
    const int cur = kki & 1;
    if (kki + 1 < nk) {
      stage_async(A, K, m0, (kki + 1) << 5, &As[(cur ^ 1) * 128 * LDT], tid);
      stage_async(B, K, n0, (kki + 1) << 5, &Bs[(cur ^ 1) * 128 * LDT], tid);
    }
    const float* Ac = &As[cur * 128 * LDT];
    const float* Bc = &Bs[cur * 128 * LDT];
    for (int k = 0; k < 32; k += 4) {
      v2f af[2], bf[4];
      for (int mt = 0; mt < 2; ++mt)
        af[mt] = *(const v2f*)&Ac[(wm * 32 + mt * 16 + lrow) * LDT + k + koff];
      for (int nt = 0; nt < 4; ++nt)
        bf[nt] = *(const v2f*)&Bc[(wn * 64 + nt * 16 + lrow) * LDT + k + koff];
      for (int mt = 0; mt < 2; ++mt)
        for (int nt = 0; nt < 4; ++nt)
          acc[mt][nt] = wmma_f32(af[mt], bf[nt], acc[mt][nt]);
    }
    wait_async0();   // prefetched next chunk landed
    __syncthreads(); // everyone done reading cur + everyone's prefetch visible
  }

  for (int mt = 0; mt < 2; ++mt) {
    for (int nt = 0; nt < 4; ++nt) {
      int col = n0 + wn * 64 + nt * 16 + lrow;
      float bv = bias ? bias[col] : 0.f;
      for (int v = 0; v < 8; ++v) {
        int row = m0 + wm * 32 + mt * 16 + v + mhalf;
        C[(size_t)row * N + col] = acc[mt][nt][v] + bv;
      }
    }
  }
}

// ---------------------------------------------------------------------------
// In-place L2 row normalization, D fixed at 512 (one block per row).
// ---------------------------------------------------------------------------
__global__ __launch_bounds__(256)
void l2norm_rows_kernel(float* __restrict__ X) {
  __shared__ float red[256];
  const size_t row = blockIdx.x;
  const int t = threadIdx.x;
  float x0 = X[row * 512 + t];
  float x1 = X[row * 512 + t + 256];
  red[t] = x0 * x0 + x1 * x1;
  __syncthreads();
  for (int s = 128; s > 0; s >>= 1) {
    if (t < s) red[t] += red[t + s];
    __syncthreads();
  }
  float inv = 1.f / fmaxf(sqrtf(red[0]), 1e-12f);
  X[row * 512 + t]       = x0 * inv;
  X[row * 512 + t + 256] = x1 * inv;
}

// ---------------------------------------------------------------------------
// inv_b[c] = 1 / max(sum_n exc[n][c], 1)   (single block; 8 partials per col)
// ---------------------------------------------------------------------------
__global__ __launch_bounds__(256)
void colsum_inv_kernel(const float* __restrict__ exc, float* __restrict__ inv_b, int N) {
  __shared__ float red[256];
  const int t = threadIdx.x;
  const int c = t & 31, g = t >> 5;
  float s = 0.f;
  if (c < 28)
    for (int n = g; n < N; n += 8) s += exc[(size_t)n * 28 + c];
  red[t] = s;
  __syncthreads();
  if (t < 32) {
    float tot = 0.f;
    for (int i = 0; i < 8; ++i) tot += red[t + i * 32];
    inv_b[t] = (t < 28) ? 1.f / fmaxf(tot, 1.f) : 1.f;
  }
}

// ---------------------------------------------------------------------------
// Fused:  S = fn_block(128xK) @ efn_tile(128xK)^T  (fp32 WMMA, regs, dbl-buf)
//         a = S^3 (elementwise, regs)  -> a_s in LDS (relayout to A-frag order)
//         echo_partial += a(128x128) @ exc_tile(128x32)  (fp32 WMMA, persistent)
// Grid: (NSPLIT, B/128).  s never touches HBM.
// ---------------------------------------------------------------------------
__global__ __launch_bounds__(256)
void simecho_kernel(const float* __restrict__ fn, const float* __restrict__ efn,
                    const float* __restrict__ exc, float* __restrict__ partial,
                    int B, int N, int K /*D_red*/) {
  __shared__ float As[2 * 128 * LDT];
  __shared__ float Bs[2 * 128 * LDT];
  __shared__ float Es[128 * LDT];     // exc tile [n][c], c padded to 32
  __shared__ float a_s[128 * LDA_S];  // cubed s tile [m][n]

  const int tid  = threadIdx.x;
  const int lane = tid & 31;
  const int w    = tid >> 5;
  const int wm   = w & 3;
  const int wn   = w >> 2;
  const int lrow = lane & 15;
  const int koff = (lane >> 4) << 1;
  const int mhalf = (lane >> 4) << 3;

  const int m0 = blockIdx.y * 128;
  const int ntiles_per = (N / 128) / NSPLIT;
  const int nt0 = blockIdx.x * ntiles_per;
  const int nk = K >> 5;

  // Zero-init Es pad columns (28..31) once; async loads never touch them.
  if (tid < 128) *(float4*)&Es[tid * LDT + 28] = make_float4(0.f, 0.f, 0.f, 0.f);

  v8f eacc[2];
  eacc[0] = v8f_zero(); eacc[1] = v8f_zero();

  for (int nti = nt0; nti < nt0 + ntiles_per; ++nti) {
    const int n0 = nti * 128;

    // exc tile for this n0: issue early, overlaps the whole K loop.
    // (prev echo phase's trailing barrier guarantees Es is free)
    for (int i = 0; i < 4; ++i) {
      int s = tid + i * 256;
      int r = s >> 3, c = (s & 7) << 2;
      if (c < 28)
        async_ld_b128(&Es[r * LDT + c], &exc[(size_t)(n0 + r) * 28 + c]);
    }

    // ---- main GEMM: S tile in registers, double-buffered K ----
    v8f acc[2][4];
    for (int i = 0; i < 2; ++i) for (int j = 0; j < 4; ++j) acc[i][j] = v8f_zero();

    stage_async(fn,  K, m0, 0, &As[0], tid);
    stage_async(efn, K, n0, 0, &Bs[0], tid);
    wait_async0();
    __syncthreads();

    for (int kki = 0; kki < nk; ++kki) {
      const int cur = kki & 1;
      if (kki + 1 < nk) {
        stage_async(fn,  K, m0, (kki + 1) << 5, &As[(cur ^ 1) * 128 * LDT], tid);
        stage_async(efn, K, n0, (kki + 1) << 5, &Bs[(cur ^ 1) * 128 * LDT], tid);
      }
      const float* Ac = &As[cur * 128 * LDT];
      const float* Bc = &Bs[cur * 128 * LDT];
      for (int k = 0; k < 32; k += 4) {
        v2f af[2], bf[4];
        for (int mt = 0; mt < 2; ++mt)
          af[mt] = *(const v2f*)&Ac[(wm * 32 + mt * 16 + lrow) * LDT + k + koff];
        for (int nt = 0; nt < 4; ++nt)
          bf[nt] = *(const v2f*)&Bc[(wn * 64 + nt * 16 + lrow) * LDT + k + koff];
        for (int mt = 0; mt < 2; ++mt)
          for (int nt = 0; nt < 4; ++nt)
            acc[mt][nt] = wmma_f32(af[mt], bf[nt], acc[mt][nt]);
      }
      wait_async0();
      __syncthreads();
    }

    // ---- power-sign activation (p=3 => s^3) + relayout into LDS ----
    for (int mt = 0; mt < 2; ++mt)
      for (int nt = 0; nt < 4; ++nt)
        for (int v = 0; v < 8; ++v) {
          int ml = wm * 32 + mt * 16 + v + mhalf;
          int nl = wn * 64 + nt * 16 + lrow;
          float sv = acc[mt][nt][v];
          a_s[ml * LDA_S + nl] = sv * sv * sv;
        }
    __syncthreads();   // a_s + Es published (Es async already drained in K loop)

    // ---- echo GEMM: wave w owns rows [w*16, w*16+16), K' = 128 ----
    for (int kp = 0; kp < 128; kp += 4) {
      v2f af = *(const v2f*)&a_s[(w * 16 + lrow) * LDA_S + kp + koff];
      for (int ct = 0; ct < 2; ++ct) {
        v2f bf;
        bf.x = Es[(kp + koff)     * LDT + ct * 16 + lrow];
        bf.y = Es[(kp + koff + 1) * LDT + ct * 16 + lrow];
        eacc[ct] = wmma_f32(af, bf, eacc[ct]);
      }
    }
    __syncthreads();   // echo reads done before next iter overwrites Es/a_s
  }

  // ---- write deterministic partial sums (pre-scale, pre-clamp) ----
  for (int ct = 0; ct < 2; ++ct) {
    int c = ct * 16 + lrow;
    if (c < 28) {
      for (int v = 0; v < 8; ++v) {
        int m = m0 + w * 16 + v + mhalf;
        partial[((size_t)blockIdx.x * B + m) * 28 + c] = eacc[ct][v];
      }
    }
  }
}

// ---------------------------------------------------------------------------
// out = clip( (sum_split partial) * inv_b[c], 0, 1 )
// ---------------------------------------------------------------------------
__global__ __launch_bounds__(256)
void finalize_kernel(const float* __restrict__ partial, const float* __restrict__ inv_b,
                     float* __restrict__ out, int B) {
  int idx = blockIdx.x * 256 + threadIdx.x;
  if (idx >= B * 28) return;
  int c = idx % 28;
  float s = 0.f;
  for (int i = 0; i < NSPLIT; ++i) s += partial[(size_t)i * B * 28 + idx];
  float v = s * inv_b[c];
  out[idx] = fminf(fmaxf(v, 0.f), 1.f);
}

extern "C" void kernel_launch(void* const* d_in, const int* in_sizes, int n_in,
                              void* d_out, int out_size, void* d_ws, size_t ws_size,
                              hipStream_t stream) {
  const float* features    = (const float*)d_in[0];  // [B, D_IN]
  const float* ex_features = (const float*)d_in[1];  // [N, D_IN]
  const float* g_w         = (const float*)d_in[2];  // [D_RED, D_IN]
  const float* g_b         = (const float*)d_in[3];  // [D_RED]
  const float* ex_classes  = (const float*)d_in[4];  // [N, 28]

  const int DR  = in_sizes[3];                 // 512
  const int DIN = in_sizes[2] / DR;            // 1024
  const int B   = in_sizes[0] / DIN;           // 8192
  const int N   = in_sizes[1] / DIN;           // 16384

  float* fn      = (float*)d_ws;               // [B, DR]
  float* efn     = fn + (size_t)B * DR;        // [N, DR] (contiguous after fn)
  float* inv_b   = efn + (size_t)N * DR;       // [32]
  float* partial = inv_b + 32;                 // [NSPLIT, B, 28]

  gemm_nt_bias_kernel<<<dim3(DR / 128, B / 128), 256, 0, stream>>>(
      features, g_w, g_b, fn, B, DR, DIN);
  gemm_nt_bias_kernel<<<dim3(DR / 128, N / 128), 256, 0, stream>>>(
      ex_features, g_w, g_b, efn, N, DR, DIN);

  l2norm_rows_kernel<<<B + N, 256, 0, stream>>>(fn);

  colsum_inv_kernel<<<1, 256, 0, stream>>>(ex_classes, inv_b, N);

  simecho_kernel<<<dim3(NSPLIT, B / 128), 256, 0, stream>>>(
      fn, efn, ex_classes, partial, B, N, DR);

  finalize_kernel<<<(B * 28 + 255) / 256, 256, 0, stream>>>(
      partial, inv_b, (float*)d_out, B);
}